// Router_11081015623717
// MI455X (gfx1250) — compile-verified
//
#include <hip/hip_runtime.h>
#include <math.h>

typedef __attribute__((ext_vector_type(2))) float v2f;
typedef __attribute__((ext_vector_type(8))) float v8f;

#define T_TOK        8192
#define D_DIM        4096
#define E_EXP        64
#define TOPK         8
#define M_PER_BLOCK  32
#define LDS_STRIDE   65   // pad 16x-rows to avoid LDS bank conflicts

__global__ __launch_bounds__(128) void router_fused_kernel(
    const float* __restrict__ x,      // [T, D] row-major
    const float* __restrict__ wde,    // [D, E] row-major
    float* __restrict__ out_w,        // [T, TOPK]
    int*   __restrict__ out_i)        // [T, TOPK]
{
    __shared__ float lds[M_PER_BLOCK * LDS_STRIDE];

    const int tid   = threadIdx.x;
    const int wid   = tid >> 5;       // wave id 0..3 -> expert slice
    const int lane  = tid & 31;
    const int half  = lane >> 4;      // 0: lanes 0-15, 1: lanes 16-31
    const int l16   = lane & 15;
    const int kq    = half << 1;      // K sub-offset 0 or 2 (A and B layouts)
    const int tok0  = blockIdx.x * M_PER_BLOCK;
    const int nbase = wid << 4;       // expert column base for this wave

    // A: lane holds x[m][k0+kq], x[m][k0+kq+1]  (contiguous -> b64 loads)
    const float* a0p = x + (size_t)(tok0 + l16) * D_DIM + kq;
    const float* a1p = a0p + (size_t)16 * D_DIM;
    // B: lane holds W[k0+kq][n], W[k0+kq+1][n]  (stride E between the two)
    const float* bp  = wde + (size_t)kq * E_EXP + (size_t)(nbase + l16);

    v8f acc0 = {};   // C tile for tokens tok0..tok0+15
    v8f acc1 = {};   // C tile for tokens tok0+16..tok0+31

    #pragma unroll 2
    for (int k0 = 0; k0 < D_DIM; k0 += 8) {
        // two K=4 steps per iteration; both M tiles share the B fragments
        v2f a0a = __builtin_nontemporal_load((const v2f*)(a0p + k0));
        v2f a1a = __builtin_nontemporal_load((const v2f*)(a1p + k0));
        v2f a0b = __builtin_nontemporal_load((const v2f*)(a0p + k0 + 4));
        v2f a1b = __builtin_nontemporal_load((const v2f*)(a1p + k0 + 4));

        v2f b0, b1;
        b0.x = bp[0];
        b0.y = bp[E_EXP];
        b1.x = bp[4 * E_EXP];
        b1.y = bp[5 * E_EXP];
        bp += 8 * E_EXP;

        acc0 = __builtin_amdgcn_wmma_f32_16x16x4_f32(false, a0a, false, b0,
                                                     (short)0, acc0, false, false);
        acc1 = __builtin_amdgcn_wmma_f32_16x16x4_f32(false, a1a, false, b0,
                                                     (short)0, acc1, false, false);
        acc0 = __builtin_amdgcn_wmma_f32_16x16x4_f32(false, a0b, false, b1,
                                                     (short)0, acc0, false, false);
        acc1 = __builtin_amdgcn_wmma_f32_16x16x4_f32(false, a1b, false, b1,
                                                     (short)0, acc1, false, false);
    }

    // C layout: VGPR r -> row (half*8 + r), col = nbase + l16
    const int mrow = half << 3;
    const int col  = nbase + l16;
    #pragma unroll
    for (int r = 0; r < 8; ++r) {
        lds[(mrow + r)      * LDS_STRIDE + col] = acc0[r];
        lds[(16 + mrow + r) * LDS_STRIDE + col] = acc1[r];
    }
    __syncthreads();

    // Wave 0: one lane per token -> top-8 (stable, lowest index wins ties) + softmax
    if (tid < M_PER_BLOCK) {
        const int m = tid;
        const float* row = &lds[m * LDS_STRIDE];

        float sv[TOPK];
        int   si[TOPK];
        #pragma unroll
        for (int t = 0; t < TOPK; ++t) { sv[t] = -INFINITY; si[t] = -1; }

        #pragma unroll 4
        for (int e = 0; e < E_EXP; ++e) {
            float nv = row[e];
            int   ni = e;
            #pragma unroll
            for (int t = 0; t < TOPK; ++t) {
                const bool  sw = nv > sv[t];   // strict: equal keeps earlier index first
                const float tv = sv[t];
                const int   ti = si[t];
                sv[t] = sw ? nv : tv;
                si[t] = sw ? ni : ti;
                nv    = sw ? tv : nv;
                ni    = sw ? ti : ni;
            }
        }

        const float mx = sv[0];
        float ev[TOPK];
        float sum = 0.0f;
        #pragma unroll
        for (int t = 0; t < TOPK; ++t) {
            ev[t] = __expf(sv[t] - mx);
            sum  += ev[t];
        }
        const float inv = 1.0f / sum;

        const int token = tok0 + m;
        #pragma unroll
        for (int t = 0; t < TOPK; ++t) {
            out_w[(size_t)token * TOPK + t] = ev[t] * inv;
            out_i[(size_t)token * TOPK + t] = si[t];
        }
    }
}

extern "C" void kernel_launch(void* const* d_in, const int* in_sizes, int n_in,
                              void* d_out, int out_size, void* d_ws, size_t ws_size,
                              hipStream_t stream) {
    (void)in_sizes; (void)n_in; (void)d_ws; (void)ws_size; (void)out_size;

    const float* x   = (const float*)d_in[0];   // [T, D] f32
    const float* wde = (const float*)d_in[1];   // [D, E] f32
    // d_in[2] = num_experts_per_tok (== 8, baked in as TOPK)

    float* out_w = (float*)d_out;                                  // [T, 8] f32
    int*   out_i = (int*)((float*)d_out + (size_t)T_TOK * TOPK);   // [T, 8] i32 (bit-stored)

    dim3 grid(T_TOK / M_PER_BLOCK);   // 256 blocks
    dim3 block(128);                  // 4 waves of 32
    router_fused_kernel<<<grid, block, 0, stream>>>(x, wde, out_w, out_i);
}